// SUPREME_25065429139537
// MI455X (gfx1250) — compile-verified
//
#include <hip/hip_runtime.h>

typedef __attribute__((ext_vector_type(2))) float v2f;
typedef __attribute__((ext_vector_type(8))) float v8f;

// ---------------- degree / norm ----------------

__global__ void k_deg(const long long* __restrict__ dst, float* __restrict__ deg,
                      long long E) {
  long long t = (long long)blockIdx.x * blockDim.x + threadIdx.x;
  if (t < E) atomicAdd(&deg[dst[t]], 1.0f);
}

__global__ void k_rsqrt(float* __restrict__ d, int N) {
  int i = blockIdx.x * blockDim.x + threadIdx.x;
  if (i < N) d[i] = rsqrtf(d[i] + 1.0f);
}

__global__ void k_norm(const long long* __restrict__ src, const long long* __restrict__ dst,
                       const float* __restrict__ dis, float* __restrict__ norm,
                       long long E) {
  long long t = (long long)blockIdx.x * blockDim.x + threadIdx.x;
  if (t < E) norm[t] = dis[src[t]] * dis[dst[t]];
}

// ---------------- f32 WMMA GEMM: C[M,Nc] = A[M,K] * B[K,Nc] ----------------
// One 16x16 output tile per wave; V_WMMA_F32_16X16X4_F32, K stepped by 4.
// Layout per CDNA5 ISA 7.12.2 (32-bit operands, wave32):
//   A 16x4 : lane%16 = M row; VGPR v, lane-half h -> K = v + 2h
//   B 4x16 : VGPR v, lane-half h -> K row = v + 2h; lane%16 = N col
//   C 16x16: VGPR r, lane-half h -> M = r + 8h; lane%16 = N col
__global__ void k_gemm_wmma(const float* __restrict__ A, const float* __restrict__ B,
                            float* __restrict__ C, int M, int K, int Nc) {
  int wave = (int)(((long long)blockIdx.x * blockDim.x + threadIdx.x) >> 5);
  int lane = threadIdx.x & 31;
  int ntiles = Nc >> 4;
  int mt = wave / ntiles;
  int nt = wave - mt * ntiles;
  if (mt * 16 >= M) return;            // wave-uniform exit: EXEC stays all-ones
  int half = lane >> 4;
  int l16  = lane & 15;
  int row = mt * 16 + l16;             // A row for this lane
  int col = nt * 16 + l16;             // B/C column for this lane

  v8f acc = {};
  for (int kk = 0; kk < K; kk += 4) {
    int ka = kk + half * 2;
    v2f a, b;
    a.x = A[(long long)row * K + ka];
    a.y = A[(long long)row * K + ka + 1];
    b.x = B[(long long)ka * Nc + col];
    b.y = B[(long long)(ka + 1) * Nc + col];
    acc = __builtin_amdgcn_wmma_f32_16x16x4_f32(
        /*neg_a=*/false, a, /*neg_b=*/false, b,
        /*c_mod=*/(short)0, acc, /*reuse_a=*/false, /*reuse_b=*/false);
  }
#pragma unroll
  for (int r = 0; r < 8; ++r) {
    int m = mt * 16 + half * 8 + r;
    C[(long long)m * Nc + col] = acc[r];
  }
}

// ---------------- edge aggregation: agg[dst,f] += h[src,f] * norm[e] ----------------
// One thread per (edge, feature); consecutive threads share an edge so the
// h[src] gather and the atomic scatter are both coalesced 128B per wave.
__global__ void k_edge_agg(const float* __restrict__ h, const long long* __restrict__ src,
                           const long long* __restrict__ dst, const float* __restrict__ norm,
                           float* __restrict__ agg, long long E, int logF) {
  long long t = (long long)blockIdx.x * blockDim.x + threadIdx.x;
  long long e = t >> logF;
  if (e >= E) return;
  int f = (int)(t & ((1 << logF) - 1));
  long long s = src[e], d = dst[e];
  atomicAdd(&agg[(d << logF) + f], h[(s << logF) + f] * norm[e]);
}

// ---------------- self-loop + bias (+ReLU), in place ----------------
__global__ void k_post(float* __restrict__ agg, const float* __restrict__ h,
                       const float* __restrict__ dis, const float* __restrict__ bias,
                       long long N, int logF, int relu) {
  long long t = (long long)blockIdx.x * blockDim.x + threadIdx.x;
  long long total = N << logF;
  if (t >= total) return;
  long long i = t >> logF;
  int f = (int)(t & ((1 << logF) - 1));
  float di = dis[i];
  float v = agg[t] + h[t] * di * di + bias[f];
  if (relu) v = fmaxf(v, 0.0f);
  agg[t] = v;
}

// ---------------- launch ----------------

extern "C" void kernel_launch(void* const* d_in, const int* in_sizes, int n_in,
                              void* d_out, int out_size, void* d_ws, size_t ws_size,
                              hipStream_t stream) {
  const float*     x   = (const float*)d_in[0];
  const long long* ei  = (const long long*)d_in[1];   // int64 [2,E]
  const float*     W1  = (const float*)d_in[2];
  const float*     b1  = (const float*)d_in[3];
  const float*     W2  = (const float*)d_in[4];
  const float*     b2  = (const float*)d_in[5];
  float*           out = (float*)d_out;

  const int HID  = in_sizes[3];              // 64
  const int OUTF = in_sizes[5];              // 32
  const int INF  = in_sizes[2] / HID;        // 64
  const long long N = in_sizes[0] / INF;     // 100000 (divisible by 16)
  const long long E = in_sizes[1] / 2;       // 1.6M

  const long long* src = ei;
  const long long* dst = ei + E;

  int logH = 0; while ((1 << logH) < HID)  ++logH;   // 6
  int logO = 0; while ((1 << logO) < OUTF) ++logO;   // 5

  // workspace layout (floats)
  float* ws   = (float*)d_ws;
  float* dis  = ws;                      size_t o = (size_t)N;
  float* norm = ws + o;                  o += (size_t)E;
  float* h1   = ws + o;                  o += (size_t)N * HID;
  float* agg1 = ws + o;                  o += (size_t)N * HID;
  float* h2   = ws + o;                  /* o += N*OUTF */

  const int B = 256;  // 8 wave32 waves per block

  // zero accumulators every call (graph-replay safe)
  hipMemsetAsync(dis,  0, (size_t)N * sizeof(float), stream);
  hipMemsetAsync(agg1, 0, (size_t)N * HID * sizeof(float), stream);
  hipMemsetAsync(out,  0, (size_t)N * OUTF * sizeof(float), stream);

  // degree -> dis -> per-edge norm (shared by both layers)
  k_deg  <<<(int)((E + B - 1) / B), B, 0, stream>>>(dst, dis, E);
  k_rsqrt<<<(int)((N + B - 1) / B), B, 0, stream>>>(dis, (int)N);
  k_norm <<<(int)((E + B - 1) / B), B, 0, stream>>>(src, dst, dis, norm, E);

  // ---- layer 1 ----
  {
    long long waves = (N / 16) * (HID / 16);
    long long thr = waves * 32;
    k_gemm_wmma<<<(int)((thr + B - 1) / B), B, 0, stream>>>(x, W1, h1, (int)N, INF, HID);
    long long tot = E << logH;
    k_edge_agg<<<(int)((tot + B - 1) / B), B, 0, stream>>>(h1, src, dst, norm, agg1, E, logH);
    long long nt = N << logH;
    k_post<<<(int)((nt + B - 1) / B), B, 0, stream>>>(agg1, h1, dis, b1, N, logH, 1);
  }

  // ---- layer 2 ----
  {
    long long waves = (N / 16) * (OUTF / 16);
    long long thr = waves * 32;
    k_gemm_wmma<<<(int)((thr + B - 1) / B), B, 0, stream>>>(agg1, W2, h2, (int)N, HID, OUTF);
    long long tot = E << logO;
    k_edge_agg<<<(int)((tot + B - 1) / B), B, 0, stream>>>(h2, src, dst, norm, out, E, logO);
    long long nt = N << logO;
    k_post<<<(int)((nt + B - 1) / B), B, 0, stream>>>(out, h2, dis, b2, N, logO, 0);
  }
}